// S7_80900003987523
// MI455X (gfx1250) — compile-verified
//
#include <hip/hip_runtime.h>

// S7/S5-style event-based SSM forward for MI455X (gfx1250, wave32, WMMA).
//
// Shapes: BATCH=4, L=32768, H=256, P=128.
// Pipeline:
//   k0: transpose tiny weight mats (B -> (H,P) planes, C -> (P,H) planes) for
//       lane-coalesced WMMA B-fragment loads.
//   k1 (WMMA f32 16x16x4): step/BuR/BuI GEMMs + ZOH discretization
//       -> ws state buffer float4 (Lam_re, Lam_im, Bu_re, Bu_im).
//   k2: chunked complex scan (chunk=128), in place -> (A_cum, x_local),
//       chunk aggregates; global_prefetch_b8 16KB ahead in the serial loop.
//   k3: exclusive scan of chunk aggregates.
//   k4 (WMMA f32 16x16x4): ys = 2*Re(xs @ C~^T) + u*D, xs = x_local +
//       A_cum*prefix reconstructed on the fly (never materialized).
//
// Workspace: B*L*P*16 + B*NC*P*16 + 4*H*P*4  ~= 258 MB.

typedef float v2f __attribute__((ext_vector_type(2)));
typedef float v8f __attribute__((ext_vector_type(8)));

#define BATCH  4
#define SEQL   32768
#define HDIM   256
#define PDIM   128
#define CHUNK  128
#define NCHUNK (SEQL / CHUNK)   // 256

static __device__ __forceinline__ v8f wmma_f32x4(v2f a, v2f b, v8f c) {
    // D(16x16,f32) = A(16x4,f32) * B(4x16,f32) + C
    // 8 args: (neg_a, A, neg_b, B, c_mod, C, reuse_a, reuse_b); NEG must be 0 for F32.
    return __builtin_amdgcn_wmma_f32_16x16x4_f32(
        false, a, false, b, (short)0, c, false, false);
}

// ---------------------------------------------------------------------------
// Kernel 0: weight transposes (tiny, 512 KB of ws planes).
//   BrT/BiT: (H,P)  BrT[h*P+p] = B[p,h,re/im]   (k=h rows, n=p lane-consecutive)
//   CrT/CiT: (P,H)  CrT[p*H+h] = C[h,p,re/im]   (k=p rows, n=h lane-consecutive)
// ---------------------------------------------------------------------------
__global__ void __launch_bounds__(256)
s7_xpose(const float* __restrict__ Bmat,   // (P,H,2)
         const float* __restrict__ Cmat,   // (H,P,2)
         float* __restrict__ BrT, float* __restrict__ BiT,
         float* __restrict__ CrT, float* __restrict__ CiT)
{
    const int i = blockIdx.x * 256 + threadIdx.x;   // 0 .. H*P-1
    const int h = i / PDIM;
    const int p = i % PDIM;
    BrT[(size_t)h * PDIM + p] = Bmat[((size_t)p * HDIM + h) * 2 + 0];
    BiT[(size_t)h * PDIM + p] = Bmat[((size_t)p * HDIM + h) * 2 + 1];
    CrT[(size_t)p * HDIM + h] = Cmat[((size_t)h * PDIM + p) * 2 + 0];
    CiT[(size_t)p * HDIM + h] = Cmat[((size_t)h * PDIM + p) * 2 + 1];
}

// ---------------------------------------------------------------------------
// Kernel 1: fused triple GEMM (step, BuR, BuI) + ZOH discretization.
// One wave computes one 16(L) x 16(P) output tile; 8 waves per block cover
// 128 consecutive rows. Grid: (L/128, P/16, BATCH).
// ---------------------------------------------------------------------------
__global__ void __launch_bounds__(256)
s7_frontend(const float* __restrict__ u,      // (B,L,H)
            const float* __restrict__ td,     // (B,L)
            const float* __restrict__ Lre,    // (P)
            const float* __restrict__ Lim,    // (P)
            const float* __restrict__ BrT,    // (H,P)
            const float* __restrict__ BiT,    // (H,P)
            const float* __restrict__ Wstep,  // (H,P)
            const float* __restrict__ bstep,  // (P)
            float4* __restrict__ st)          // (B,L,P): (Lam_re,Lam_im,Bu_re,Bu_im)
{
    const int lane = threadIdx.x & 31;
    const int wave = threadIdx.x >> 5;
    const int b    = blockIdx.z;
    const int row0 = (blockIdx.x * 8 + wave) * 16;
    const int col0 = blockIdx.y * 16;
    const int m    = lane & 15;   // A-row / B-col lane index
    const int kp   = lane >> 4;   // K-pair select (0 or 1)

    v8f accS = {};  // step_pre
    v8f accR = {};  // u @ B_re^T
    v8f accI = {};  // u @ B_im^T

    const size_t uRow = ((size_t)b * SEQL + row0 + m) * HDIM;
    const int    nCol = col0 + m;

    for (int k0 = 0; k0 < HDIM; k0 += 4) {
        const int ka = k0 + 2 * kp;
        const int kb = ka + 1;
        v2f a;  a.x  = u[uRow + ka];                      a.y  = u[uRow + kb];
        v2f w;  w.x  = Wstep[(size_t)ka * PDIM + nCol];   w.y  = Wstep[(size_t)kb * PDIM + nCol];
        v2f br; br.x = BrT[(size_t)ka * PDIM + nCol];     br.y = BrT[(size_t)kb * PDIM + nCol];
        v2f bi; bi.x = BiT[(size_t)ka * PDIM + nCol];     bi.y = BiT[(size_t)kb * PDIM + nCol];
        accS = wmma_f32x4(a, w,  accS);
        accR = wmma_f32x4(a, br, accR);
        accI = wmma_f32x4(a, bi, accI);
    }

    // Elementwise ZOH stage. Lane owns column p = col0+m; rows m_out = v + 8*kp.
    const int   p      = nCol;
    const float lre    = Lre[p];
    const float lim    = Lim[p];
    const float invden = 1.0f / (lre * lre + lim * lim);
    const float bstp   = bstep[p];

#pragma unroll
    for (int v = 0; v < 8; ++v) {
        const int   l   = row0 + v + 8 * kp;
        const float tdl = td[(size_t)b * SEQL + l];

        float x  = accS[v] + bstp;
        float sp = (x > 20.0f) ? x : log1pf(__expf(x));   // softplus
        float dl = sp * tdl;                              // Delta

        float e  = __expf(lre * dl);
        float sc, cc;
        __sincosf(lim * dl, &sc, &cc);
        float cr = e * cc;                                // Lambda_bar (re, im)
        float ci = e * sc;

        // gamma = (Lambda_bar - 1) / Lambda  (complex divide via conj)
        float gr = ((cr - 1.0f) * lre + ci * lim) * invden;
        float gi = (ci * lre - (cr - 1.0f) * lim) * invden;

        float ur = accR[v], ui = accI[v];
        float vr = gr * ur - gi * ui;                     // Bu = gamma * (u @ B~^T)
        float vi = gr * ui + gi * ur;

        st[((size_t)b * SEQL + l) * PDIM + p] = make_float4(cr, ci, vr, vi);
    }
}

// ---------------------------------------------------------------------------
// Kernel 2: per-chunk sequential complex scan, in place (single b128 per step).
// st -> (A_cum_re, A_cum_im, x_local_re, x_local_im).
// One thread per p channel; block=(P), grid=(NCHUNK, BATCH) -> 131072 lanes.
// global_prefetch_b8 issued 8 steps (16 KB) ahead of the serial recurrence.
// ---------------------------------------------------------------------------
__global__ void __launch_bounds__(PDIM)
s7_scan_chunk(float4* __restrict__ st,
              float4* __restrict__ agg)     // (B,NCHUNK,P): (A_re,A_im,x_re,x_im)
{
    const int p     = threadIdx.x;
    const int chunk = blockIdx.x;
    const int b     = blockIdx.y;
    size_t idx = ((size_t)b * SEQL + (size_t)chunk * CHUNK) * PDIM + p;

    float Ar = 1.0f, Ai = 0.0f, xr = 0.0f, xi = 0.0f;
    for (int t = 0; t < CHUNK; ++t, idx += PDIM) {
        if ((t & 7) == 0)
            __builtin_prefetch(st + idx + 8 * PDIM, 0, 3);   // -> global_prefetch_b8
        const float4 s = st[idx];
        const float nxr = s.x * xr - s.y * xi + s.z;   // x = A*x + Bu
        const float nxi = s.x * xi + s.y * xr + s.w;
        xr = nxr; xi = nxi;
        const float nAr = s.x * Ar - s.y * Ai;         // Acum *= A
        const float nAi = s.x * Ai + s.y * Ar;
        Ar = nAr; Ai = nAi;
        st[idx] = make_float4(Ar, Ai, xr, xi);
    }
    agg[((size_t)b * NCHUNK + chunk) * PDIM + p] = make_float4(Ar, Ai, xr, xi);
}

// ---------------------------------------------------------------------------
// Kernel 3: exclusive scan over the NCHUNK chunk aggregates per (batch,p).
// After this, agg[c] holds the state *entering* chunk c (identity for c=0).
// ---------------------------------------------------------------------------
__global__ void __launch_bounds__(PDIM)
s7_scan_agg(float4* __restrict__ agg)
{
    const int p = threadIdx.x;
    const int b = blockIdx.x;
    float pAr = 1.0f, pAi = 0.0f, pxr = 0.0f, pxi = 0.0f;
    for (int c = 0; c < NCHUNK; ++c) {
        const size_t i = ((size_t)b * NCHUNK + c) * PDIM + p;
        const float4 t = agg[i];
        agg[i] = make_float4(pAr, pAi, pxr, pxi);          // exclusive prefix
        const float nAr = t.x * pAr - t.y * pAi;           // combine
        const float nAi = t.x * pAi + t.y * pAr;
        const float nxr = t.x * pxr - t.y * pxi + t.z;
        const float nxi = t.x * pxi + t.y * pxr + t.w;
        pAr = nAr; pAi = nAi; pxr = nxr; pxi = nxi;
    }
}

// ---------------------------------------------------------------------------
// Kernel 4: output GEMM  ys = 2*Re(xs @ C~^T) + u*D, with
// xs = x_local + A_cum * prefix_x reconstructed on the fly (saves 256 MB of
// xs round-trip). Re(xs@C^T) = XR@Cr^T - XI@Ci^T; XI is negated in VALU since
// F32 WMMA forbids A/B NEG modifiers. Grid: (L/128, H/16, BATCH).
// ---------------------------------------------------------------------------
__global__ void __launch_bounds__(256)
s7_backend(const float* __restrict__ u,       // (B,L,H)
           const float* __restrict__ CrT,     // (P,H)
           const float* __restrict__ CiT,     // (P,H)
           const float* __restrict__ Dvec,    // (H)
           const float4* __restrict__ st,     // (A_cum, x_local)
           const float4* __restrict__ agg,    // prefix (z,w = x)
           float* __restrict__ out)           // (B,L,H)
{
    const int lane  = threadIdx.x & 31;
    const int wave  = threadIdx.x >> 5;
    const int b     = blockIdx.z;
    const int row0  = (blockIdx.x * 8 + wave) * 16;
    const int col0  = blockIdx.y * 16;
    const int m     = lane & 15;
    const int kp    = lane >> 4;
    const int chunk = row0 / CHUNK;   // 16-row tile never crosses a 128-row chunk

    v8f acc = {};
    const size_t xRow    = ((size_t)b * SEQL + row0 + m) * PDIM;
    const size_t aggBase = ((size_t)b * NCHUNK + chunk) * PDIM;
    const int    nCol    = col0 + m;

    for (int k0 = 0; k0 < PDIM; k0 += 4) {
        const int ka = k0 + 2 * kp;
        v2f ar, an;   // re fragment, negated-im fragment
#pragma unroll
        for (int j = 0; j < 2; ++j) {
            const int    pk = ka + j;
            const float4 s  = st[xRow + pk];               // (Acum, x_local)
            const float4 pf = agg[aggBase + pk];
            const float  xr = s.z + s.x * pf.z - s.y * pf.w;
            const float  xi = s.w + s.x * pf.w + s.y * pf.z;
            if (j == 0) { ar.x = xr; an.x = -xi; }
            else        { ar.y = xr; an.y = -xi; }
        }
        v2f cr, ci;
        cr.x = CrT[(size_t)ka * HDIM + nCol];
        cr.y = CrT[(size_t)(ka + 1) * HDIM + nCol];
        ci.x = CiT[(size_t)ka * HDIM + nCol];
        ci.y = CiT[(size_t)(ka + 1) * HDIM + nCol];
        acc = wmma_f32x4(ar, cr, acc);   // + XR @ Cr^T
        acc = wmma_f32x4(an, ci, acc);   // - XI @ Ci^T
    }

    const int   h  = nCol;
    const float dh = Dvec[h];
#pragma unroll
    for (int v = 0; v < 8; ++v) {
        const int    l = row0 + v + 8 * kp;
        const size_t o = ((size_t)b * SEQL + l) * HDIM + h;
        out[o] = 2.0f * acc[v] + u[o] * dh;
    }
}

// ---------------------------------------------------------------------------
extern "C" void kernel_launch(void* const* d_in, const int* in_sizes, int n_in,
                              void* d_out, int out_size, void* d_ws, size_t ws_size,
                              hipStream_t stream) {
    (void)in_sizes; (void)n_in; (void)out_size; (void)ws_size;

    const float* u     = (const float*)d_in[0];  // (B,L,H)
    const float* td    = (const float*)d_in[1];  // (B,L)
    const float* Lre   = (const float*)d_in[2];  // (P)
    const float* Lim   = (const float*)d_in[3];  // (P)
    const float* Bmat  = (const float*)d_in[4];  // (P,H,2)
    const float* Cmat  = (const float*)d_in[5];  // (H,P,2)
    const float* Dvec  = (const float*)d_in[6];  // (H)
    const float* Wstep = (const float*)d_in[7];  // (H,P)
    const float* bstep = (const float*)d_in[8];  // (P)
    float* out = (float*)d_out;

    // Workspace partition (~258 MB needed):
    const size_t NLP  = (size_t)BATCH * SEQL * PDIM;       // 16,777,216
    const size_t NAGG = (size_t)BATCH * NCHUNK * PDIM;     // 131,072
    float4* st  = (float4*)d_ws;                           // 256 MB
    float4* agg = st + NLP;                                // 2 MB
    float*  BrT = (float*)(agg + NAGG);                    // 4 x 128 KB planes
    float*  BiT = BrT + (size_t)HDIM * PDIM;
    float*  CrT = BiT + (size_t)HDIM * PDIM;
    float*  CiT = CrT + (size_t)HDIM * PDIM;

    s7_xpose<<<dim3((HDIM * PDIM) / 256), 256, 0, stream>>>(Bmat, Cmat,
                                                            BrT, BiT, CrT, CiT);

    dim3 g1(SEQL / 128, PDIM / 16, BATCH);                 // (256, 8, 4)
    s7_frontend<<<g1, 256, 0, stream>>>(u, td, Lre, Lim, BrT, BiT, Wstep, bstep, st);

    dim3 g2(NCHUNK, BATCH);                                // (256, 4)
    s7_scan_chunk<<<g2, PDIM, 0, stream>>>(st, agg);

    s7_scan_agg<<<dim3(BATCH), PDIM, 0, stream>>>(agg);

    dim3 g4(SEQL / 128, HDIM / 16, BATCH);                 // (256, 16, 4)
    s7_backend<<<g4, 256, 0, stream>>>(u, CrT, CiT, Dvec, st, agg, out);
}